// AppearanceComposability_18262200943219
// MI455X (gfx1250) — compile-verified
//
#include <hip/hip_runtime.h>

// CDNA5 / gfx1250: wave32, WMMA f32 16x16x4.
typedef __attribute__((ext_vector_type(2))) float v2f;
typedef __attribute__((ext_vector_type(8))) float v8f;

#define WMMA4F32(a, b, c) \
  __builtin_amdgcn_wmma_f32_16x16x4_f32(false, (a), false, (b), (short)0, (c), false, false)

// Problem constants (from reference): B=8, C=64, H=W=32, D=64, K=7 -> L=26, O=7
#define HH 32
#define WW 32
#define DD 64
#define LWIN 26
#define OUTD 7

__global__ __launch_bounds__(256) void ac_wmma_kernel(
    const float* __restrict__ key_map,
    const float* __restrict__ query_map,
    float* __restrict__ out)
{
    const int lane  = threadIdx.x & 31;
    const int wib   = threadIdx.x >> 5;             // wave-in-block 0..7
    const int wid   = blockIdx.x * 8 + wib;         // 0..2047
    const int plane = wid >> 2;                     // b*64+c : 0..511
    const int dt    = wid & 3;                      // d-tile (16 wide)

    const int jrow  = lane & 15;                    // A-matrix row (j window idx), D-col N
    const int khalf = (lane >> 4) << 1;             // 0 (lanes 0-15) / 2 (lanes 16-31)
    const int dof   = dt * 16 + jrow;               // this lane's d column

    const float* kp = key_map + (size_t)plane * (HH * WW * DD);

    // query scalar for this lane's d column: query_map[b,c,16,16,d]
    const float q = query_map[(size_t)plane * (HH * WW * DD) + (16 * WW + 16) * DD + dof];

    // ---- A matrix: constant 0/1 window mask, rows j=0..6 valid ----
    // Layout (32-bit A 16x4): lane%16 = M(=j); VGPR0 <-> K = khalf, VGPR1 <-> K = khalf+1
    v2f A[8];
#pragma unroll
    for (int kc = 0; kc < 8; ++kc) {
        const int w0 = kc * 4 + khalf;
        const int w1 = w0 + 1;
        A[kc].x = (jrow < OUTD && w0 >= jrow && w0 <= jrow + (LWIN - 1)) ? 1.0f : 0.0f;
        A[kc].y = (jrow < OUTD && w1 >= jrow && w1 <= jrow + (LWIN - 1)) ? 1.0f : 0.0f;
    }

    // 4 independent accumulator chains (shortens WMMA->WMMA RAW chains)
    const v8f vzero = {0.f, 0.f, 0.f, 0.f, 0.f, 0.f, 0.f, 0.f};
    v8f acc0 = vzero, acc1 = vzero, acc2 = vzero, acc3 = vzero;
    v8f lowC[OUTD - 1];                              // Ccum after 1..6 rows

    // Double-buffered B row: B layout (32-bit 4x16): VGPR0 <-> K=khalf row, VGPR1 <-> K=khalf+1
    v2f Bbuf[2][8];

#define LOAD_ROW(h, buf)                                               \
    do {                                                               \
        const float* row_ = kp + (size_t)(h) * (WW * DD) + dof;        \
        _Pragma("unroll")                                              \
        for (int kc = 0; kc < 8; ++kc) {                               \
            const int w0_ = kc * 4 + khalf;                            \
            (buf)[kc].x = row_[(size_t)w0_ * DD];                      \
            (buf)[kc].y = row_[(size_t)(w0_ + 1) * DD];                \
        }                                                              \
    } while (0)

    LOAD_ROW(0, Bbuf[0]);

#pragma unroll
    for (int h = 0; h < HH; ++h) {
        v2f* cur = Bbuf[h & 1];
        if (h + 1 < HH) LOAD_ROW(h + 1, Bbuf[(h + 1) & 1]);

        // S1 accumulation: Ccum += A x X_row  (8 K-chunks of 4 over w=0..31)
        acc0 = WMMA4F32(A[0], cur[0], acc0);
        acc1 = WMMA4F32(A[1], cur[1], acc1);
        acc2 = WMMA4F32(A[2], cur[2], acc2);
        acc3 = WMMA4F32(A[3], cur[3], acc3);
        acc0 = WMMA4F32(A[4], cur[4], acc0);
        acc1 = WMMA4F32(A[5], cur[5], acc1);
        acc2 = WMMA4F32(A[6], cur[6], acc2);
        acc3 = WMMA4F32(A[7], cur[7], acc3);

        const int rowsDone = h + 1;

        // snapshot Ccum after 1..6 rows (C[i] for i=1..6)
        if (rowsDone >= 1 && rowsDone <= OUTD - 1) {
            lowC[rowsDone - 1] = (acc0 + acc1) + (acc2 + acc3);
        }

        // emit out[i] = Ccum[i+26] - Ccum[i] once row i+25 is folded in
        if (rowsDone >= LWIN) {
            const int i = rowsDone - LWIN;                  // 0..6
            v8f cum = (acc0 + acc1) + (acc2 + acc3);
            v8f res = (i == 0) ? cum : (cum - lowC[i - 1]);

            float* ob = out + (((size_t)plane * OUTD + i) * OUTD) * DD + dt * 16;
            if (lane < 16) {
#pragma unroll
                for (int r = 0; r < OUTD; ++r) {            // VGPR r holds row j=r (lanes 0-15)
                    ob[(size_t)r * DD + lane] = res[r] * q;
                }
            }
        }
    }
#undef LOAD_ROW
}

extern "C" void kernel_launch(void* const* d_in, const int* in_sizes, int n_in,
                              void* d_out, int out_size, void* d_ws, size_t ws_size,
                              hipStream_t stream)
{
    (void)in_sizes; (void)n_in; (void)out_size; (void)d_ws; (void)ws_size;
    const float* key_map   = (const float*)d_in[0];
    const float* query_map = (const float*)d_in[1];
    float* out = (float*)d_out;

    // 2048 waves total: 512 (b,c) planes x 4 d-tiles; 8 waves per 256-thread block.
    ac_wmma_kernel<<<dim3(256), dim3(256), 0, stream>>>(key_map, query_map, out);
}